// GNN_24730421690787
// MI455X (gfx1250) — compile-verified
//
#include <hip/hip_runtime.h>

#define N_NODES  50000
#define N_EDGES  800000
#define N_GRAPHS 512
#define IN_SIZE  64
#define H_SIZE   128
#define OUT_SIZE 16
#define LN_EPS   1e-5f

typedef __attribute__((ext_vector_type(2))) float v2f;
typedef __attribute__((ext_vector_type(8))) float v8f;

// ---------------- degree / dinv ----------------
__global__ void deg_init_k(float* __restrict__ deg) {
    int i = blockIdx.x * blockDim.x + threadIdx.x;
    if (i < N_NODES) deg[i] = 1.0f;   // self-loop contributes 1
}

__global__ void deg_count_k(const int* __restrict__ dst, float* __restrict__ deg) {
    int e = blockIdx.x * blockDim.x + threadIdx.x;
    if (e < N_EDGES) atomicAdd(&deg[dst[e]], 1.0f);
}

__global__ void dinv_k(float* __restrict__ deg) {
    int i = blockIdx.x * blockDim.x + threadIdx.x;
    if (i < N_NODES) deg[i] = rsqrtf(deg[i]);   // deg >= 1 always
}

// ---------------- WMMA GEMM: h' = (X @ W) * dinv[row]; acc init = h' --------
// Grid: 3125 blocks (16 rows each), 256 threads = 8 waves, wave w -> N-tile w.
// K is 64 (layer 1) or 128 (layers 2/3). Output width always H_SIZE=128.
template <int K>
__global__ __launch_bounds__(256)
void gemm_scale_k(const float* __restrict__ X, const float* __restrict__ W,
                  const float* __restrict__ dinv,
                  float* __restrict__ hp, float* __restrict__ acc) {
    const int lane   = threadIdx.x & 31;
    const int wave   = threadIdx.x >> 5;
    const int m_base = blockIdx.x << 4;
    const int n_base = wave << 4;
    const int l15    = lane & 15;
    const int kh     = (lane >> 4) << 1;          // K-half split across lane halves

    const float* __restrict__ xr = X + (size_t)(m_base + l15) * K;
    const float* __restrict__ wc = W + n_base + l15;

    v8f c = {0.f, 0.f, 0.f, 0.f, 0.f, 0.f, 0.f, 0.f};
    #pragma unroll 4
    for (int k = 0; k < K; k += 4) {
        v2f a, b;
        a.x = xr[k + kh];
        a.y = xr[k + kh + 1];
        b.x = wc[(size_t)(k + kh) * H_SIZE];
        b.y = wc[(size_t)(k + kh + 1) * H_SIZE];
        c = __builtin_amdgcn_wmma_f32_16x16x4_f32(
                /*neg_a=*/false, a, /*neg_b=*/false, b,
                /*c_mod=*/(short)0, c, /*reuse_a=*/false, /*reuse_b=*/false);
    }

    const int mh  = (lane >> 4) << 3;             // D rows: r + 8*(lane>>4)
    const int col = n_base + l15;
    #pragma unroll
    for (int r = 0; r < 8; ++r) {
        const int orow = m_base + mh + r;
        const float v  = c[r] * dinv[orow];
        hp [(size_t)orow * H_SIZE + col] = v;     // gather source
        acc[(size_t)orow * H_SIZE + col] = v;     // accumulator init == self-loop term
    }
}

// ---------------- edge scatter: acc[dst] += h'[src] ----------------
// One wave per edge: 32 lanes x float4 = 128 features, coalesced.
__global__ __launch_bounds__(256)
void scatter_k(const int* __restrict__ src, const int* __restrict__ dst,
               const float* __restrict__ hp, float* __restrict__ acc) {
    const int gid = blockIdx.x * 256 + threadIdx.x;
    const int e   = gid >> 5;
    if (e >= N_EDGES) return;
    const int f   = (gid & 31) << 2;
    const int s   = src[e];
    const int d   = dst[e];
    const float4 v = *(const float4*)(hp + (size_t)s * H_SIZE + f);
    float* p = acc + (size_t)d * H_SIZE + f;
    atomicAdd(p + 0, v.x);
    atomicAdd(p + 1, v.y);
    atomicAdd(p + 2, v.z);
    atomicAdd(p + 3, v.w);
}

// ---------------- fused dinv[dst]-scale + bias + LayerNorm + ReLU -----------
// One wave per node (lane holds 4 features), 4 nodes per 128-thread block.
__global__ __launch_bounds__(128)
void finish_ln_k(const float* __restrict__ acc, const float* __restrict__ dinv,
                 const float* __restrict__ bias, const float* __restrict__ g,
                 const float* __restrict__ be, float* __restrict__ xout) {
    const int node = blockIdx.x * 4 + (threadIdx.x >> 5);
    const int lane = threadIdx.x & 31;
    const int f    = lane << 2;

    const float4 v  = *(const float4*)(acc  + (size_t)node * H_SIZE + f);
    const float4 bb = *(const float4*)(bias + f);
    const float  di = dinv[node];
    float a0 = v.x * di + bb.x;
    float a1 = v.y * di + bb.y;
    float a2 = v.z * di + bb.z;
    float a3 = v.w * di + bb.w;

    float s = a0 + a1 + a2 + a3;
    #pragma unroll
    for (int o = 16; o > 0; o >>= 1) s += __shfl_xor(s, o, 32);
    const float mean = s * (1.0f / H_SIZE);

    const float d0 = a0 - mean, d1 = a1 - mean, d2 = a2 - mean, d3 = a3 - mean;
    float q = d0 * d0 + d1 * d1 + d2 * d2 + d3 * d3;
    #pragma unroll
    for (int o = 16; o > 0; o >>= 1) q += __shfl_xor(q, o, 32);
    const float rstd = rsqrtf(q * (1.0f / H_SIZE) + LN_EPS);

    const float4 gg = *(const float4*)(g  + f);
    const float4 eb = *(const float4*)(be + f);
    float4 r;
    r.x = fmaxf(d0 * rstd * gg.x + eb.x, 0.0f);
    r.y = fmaxf(d1 * rstd * gg.y + eb.y, 0.0f);
    r.z = fmaxf(d2 * rstd * gg.z + eb.z, 0.0f);
    r.w = fmaxf(d3 * rstd * gg.w + eb.w, 0.0f);
    *(float4*)(xout + (size_t)node * H_SIZE + f) = r;
}

// ---------------- global mean pool ----------------
__global__ void pool_init_k(float* __restrict__ pooled, float* __restrict__ counts) {
    int i = blockIdx.x * blockDim.x + threadIdx.x;
    if (i < N_GRAPHS * H_SIZE) pooled[i] = 0.0f;
    if (i < N_GRAPHS) counts[i] = 0.0f;
}

__global__ __launch_bounds__(256)
void pool_k(const float* __restrict__ x, const int* __restrict__ batch,
            float* __restrict__ pooled, float* __restrict__ counts) {
    const int gid  = blockIdx.x * 256 + threadIdx.x;
    const int node = gid >> 5;
    if (node >= N_NODES) return;
    const int lane = gid & 31;
    const int f    = lane << 2;
    const int b    = batch[node];
    const float4 v = *(const float4*)(x + (size_t)node * H_SIZE + f);
    float* p = pooled + (size_t)b * H_SIZE + f;
    atomicAdd(p + 0, v.x);
    atomicAdd(p + 1, v.y);
    atomicAdd(p + 2, v.z);
    atomicAdd(p + 3, v.w);
    if (lane == 0) atomicAdd(&counts[b], 1.0f);
}

// ---------------- final linear: (pooled/count) @ W_out + b_out --------------
__global__ void out_k(const float* __restrict__ pooled, const float* __restrict__ counts,
                      const float* __restrict__ Wout, const float* __restrict__ bout,
                      float* __restrict__ out) {
    const int gid = blockIdx.x * blockDim.x + threadIdx.x;
    if (gid >= N_GRAPHS * OUT_SIZE) return;
    const int gph = gid >> 4;
    const int o   = gid & 15;
    const float inv = 1.0f / fmaxf(counts[gph], 1.0f);
    float s = bout[o];
    const float* pr = pooled + (size_t)gph * H_SIZE;
    #pragma unroll 8
    for (int k = 0; k < H_SIZE; ++k) s = fmaf(pr[k] * inv, Wout[k * OUT_SIZE + o], s);
    out[gid] = s;
}

// ---------------- driver ----------------
extern "C" void kernel_launch(void* const* d_in, const int* in_sizes, int n_in,
                              void* d_out, int out_size, void* d_ws, size_t ws_size,
                              hipStream_t stream) {
    const float* x     = (const float*)d_in[0];
    const int*   src   = (const int*)d_in[1];               // edge_index[0]
    const int*   dst   = (const int*)d_in[1] + N_EDGES;     // edge_index[1]
    const int*   batch = (const int*)d_in[2];
    const float* W_in  = (const float*)d_in[3];
    const float* b_in  = (const float*)d_in[4];
    const float* g_in  = (const float*)d_in[5];
    const float* be_in = (const float*)d_in[6];
    const float* W1    = (const float*)d_in[7];
    const float* b1    = (const float*)d_in[8];
    const float* g1    = (const float*)d_in[9];
    const float* be1   = (const float*)d_in[10];
    const float* W2    = (const float*)d_in[11];
    const float* b2    = (const float*)d_in[12];
    const float* g2    = (const float*)d_in[13];
    const float* be2   = (const float*)d_in[14];
    const float* W_out = (const float*)d_in[15];
    const float* b_out = (const float*)d_in[16];
    float* out = (float*)d_out;

    // workspace carve-up (floats); all offsets 16B-aligned
    float* ws     = (float*)d_ws;
    float* dinv   = ws;                                   // 50000
    float* hp     = dinv + N_NODES;                       // 50000*128
    float* acc    = hp + (size_t)N_NODES * H_SIZE;        // 50000*128
    float* xbuf   = acc + (size_t)N_NODES * H_SIZE;       // 50000*128
    float* pooled = xbuf + (size_t)N_NODES * H_SIZE;      // 512*128
    float* counts = pooled + N_GRAPHS * H_SIZE;           // 512

    const int MTILES = N_NODES / 16;                      // 3125 exactly
    const int SCB    = (N_EDGES * 32) / 256;              // 100000 blocks
    const int LNB    = N_NODES / 4;                       // 12500 blocks
    const int PLB    = (N_NODES * 32 + 255) / 256;        // 6250 blocks

    // degree -> dinv
    deg_init_k <<<(N_NODES + 255) / 256, 256, 0, stream>>>(dinv);
    deg_count_k<<<(N_EDGES + 255) / 256, 256, 0, stream>>>(dst, dinv);
    dinv_k     <<<(N_NODES + 255) / 256, 256, 0, stream>>>(dinv);

    // layer 1 (K=64)
    gemm_scale_k<IN_SIZE><<<MTILES, 256, 0, stream>>>(x, W_in, dinv, hp, acc);
    scatter_k            <<<SCB,    256, 0, stream>>>(src, dst, hp, acc);
    finish_ln_k          <<<LNB,    128, 0, stream>>>(acc, dinv, b_in, g_in, be_in, xbuf);

    // layer 2 (K=128)
    gemm_scale_k<H_SIZE><<<MTILES, 256, 0, stream>>>(xbuf, W1, dinv, hp, acc);
    scatter_k           <<<SCB,    256, 0, stream>>>(src, dst, hp, acc);
    finish_ln_k         <<<LNB,    128, 0, stream>>>(acc, dinv, b1, g1, be1, xbuf);

    // layer 3 (K=128)
    gemm_scale_k<H_SIZE><<<MTILES, 256, 0, stream>>>(xbuf, W2, dinv, hp, acc);
    scatter_k           <<<SCB,    256, 0, stream>>>(src, dst, hp, acc);
    finish_ln_k         <<<LNB,    128, 0, stream>>>(acc, dinv, b2, g2, be2, xbuf);

    // pool + output
    pool_init_k<<<(N_GRAPHS * H_SIZE + 255) / 256, 256, 0, stream>>>(pooled, counts);
    pool_k     <<<PLB, 256, 0, stream>>>(xbuf, batch, pooled, counts);
    out_k      <<<(N_GRAPHS * OUT_SIZE + 255) / 256, 256, 0, stream>>>(pooled, counts, W_out, b_out, out);
}